// STSPBlock_54056458387704
// MI455X (gfx1250) — compile-verified
//
#include <hip/hip_runtime.h>
#include <math.h>

typedef __attribute__((ext_vector_type(2))) float v2f;
typedef __attribute__((ext_vector_type(8))) float v8f;

// Problem constants (from setup_inputs)
#define TT 8
#define BB 16
#define CC 16
#define CO 16
#define HH 128
#define WW 128
#define H2 64
#define W2 64
#define NM1 3

__device__ __forceinline__ float sigm(float x) { return 1.0f / (1.0f + expf(-x)); }

// ---------------------------------------------------------------------------
// LDS layouts are "channel-paired" so every WMMA operand (K-pair) is one
// aligned 8-byte LDS word -> single ds_load_b64 into an even VGPR pair.
//   weights: wl[((cp*9 + tap)*16 + co)*2 + (ci&1)],  cp = ci>>1
//   input:   xl[((cp*3 + row)*68 + col)*2 + (ci&1)]
// ---------------------------------------------------------------------------

// Kernel 1: conv0 (3x3, 16->16, SAME) + BN + ParametricLIF at 128x128.
// One wave = 16(co) x 16(pixel) tile via 36 x V_WMMA_F32_16X16X4_F32.
// Workgroup = 4 waves covering 64 consecutive columns of one row.
__global__ __launch_bounds__(128) void conv0_lif_kernel(
    const float* __restrict__ xt,     // x_t: (B,C,H,W)
    const float* __restrict__ w0,     // (16,16,3,3) OIHW
    const float* __restrict__ bng, const float* __restrict__ bnb,
    const float* __restrict__ bnm, const float* __restrict__ bnv,
    const float* __restrict__ tau0,
    float* __restrict__ v0,           // (B,16,128,128) state
    float* __restrict__ s0)           // (B,16,128,128) spikes
{
    __shared__ __align__(16) float wl[2304];          // paired weights
    __shared__ __align__(16) float xl[8 * 3 * 68 * 2]; // paired input tile

    const int tid = threadIdx.x;
    const int jg  = blockIdx.x;       // 0..1  (64-col group)
    const int h   = blockIdx.y;       // 0..127
    const int b   = blockIdx.z;       // 0..15
    const int w0col = jg * 64;

    for (int i = tid; i < 2304; i += 128) {
        int co = i / 144, r = i % 144;       // r = ci*9 + tap
        int ci = r / 9,  tap = r % 9;
        wl[(((ci >> 1) * 9 + tap) * 16 + co) * 2 + (ci & 1)] = w0[i];
    }
    for (int i = tid; i < 16 * 3 * 66; i += 128) {
        int ci = i / 198, r2 = i % 198;
        int kr = r2 / 66, col = r2 % 66;
        int gh = h + kr - 1;
        int gw = w0col - 1 + col;
        float v = 0.0f;
        if ((unsigned)gh < 128u && (unsigned)gw < 128u)
            v = xt[(((long)b * 16 + ci) * 128 + gh) * 128 + gw];
        xl[(((ci >> 1) * 3 + kr) * 68 + col) * 2 + (ci & 1)] = v;
    }
    __syncthreads();

    const int wave = tid >> 5;
    const int lane = tid & 31;
    const int pix  = lane & 15;       // N index (pixel) and M index (co) for A
    const int half = lane >> 4;

    v8f acc = {};
    #pragma unroll
    for (int kh = 0; kh < 3; ++kh)
    #pragma unroll
    for (int kw = 0; kw < 3; ++kw)
    #pragma unroll
    for (int q = 0; q < 4; ++q) {
        const int cp = 2 * q + half;                 // channel-pair index
        const v2f a  = *(const v2f*)&wl[((cp * 9 + kh * 3 + kw) * 16 + pix) * 2];
        const v2f bm = *(const v2f*)&xl[((cp * 3 + kh) * 68 + (wave * 16 + pix + kw)) * 2];
        acc = __builtin_amdgcn_wmma_f32_16x16x4_f32(
                  false, a, false, bm, (short)0, acc, false, false);
    }

    const float sg  = sigm(tau0[0]);
    const int   col = w0col + wave * 16 + pix;
    #pragma unroll
    for (int g = 0; g < 8; ++g) {
        const int co   = g + 8 * half;
        const float inv  = bng[co] * rsqrtf(bnv[co] + 1e-5f);
        const float bias = bnb[co] - bnm[co] * inv;
        const float y    = acc[g] * inv + bias;
        const long idx = (((long)b * 16 + co) * 128 + h) * 128 + col;
        const float vo = v0[idx];
        const float vn = vo + (y - vo) * sg;
        const float s  = (vn >= 1.0f) ? 1.0f : 0.0f;
        v0[idx] = vn * (1.0f - s);
        s0[idx] = s;
    }
}

// Kernel 2: 2x2 average pool: s0 (B,16,128,128) -> out0 (B,16,64,64)
__global__ void pool_kernel(const float* __restrict__ s0, float* __restrict__ out0)
{
    const int i = blockIdx.x * 256 + threadIdx.x;
    if (i >= BB * CO * H2 * W2) return;
    const int j = i & 63;
    const int r = (i >> 6) & 63;
    const int c = (i >> 12) & 15;
    const int b = i >> 16;
    const float* p = s0 + (((long)b * 16 + c) * 128 + 2 * r) * 128 + 2 * j;
    out0[i] = 0.25f * (p[0] + p[1] + p[128] + p[129]);
}

// Kernel 3: 3 inner nodes fused: conv3x3(out0/4) + BN + LIF per node,
// total = out0*ow0 + sum_n s_n*ow[n+1]. B-patch built once per tap/kchunk,
// reused for 3 WMMAs (108 WMMA per wave-tile).
__global__ __launch_bounds__(128) void inner_kernel(
    const float* __restrict__ out0,   // (B,16,64,64)
    const float* __restrict__ wc,     // (3,16,16,3,3)
    const float* __restrict__ bng, const float* __restrict__ bnb,   // (3,16)
    const float* __restrict__ bnm, const float* __restrict__ bnv,   // (3,16)
    const float* __restrict__ taus,   // (3,)
    const float* __restrict__ oww,    // (4,)
    float* __restrict__ vi,           // (3,B,16,64,64) state
    float* __restrict__ outt)         // d_out + t*B*16*64*64
{
    __shared__ __align__(16) float wl[3 * 2304];
    __shared__ __align__(16) float xl[8 * 3 * 68 * 2];

    const int tid = threadIdx.x;
    const int h = blockIdx.x;         // 0..63
    const int b = blockIdx.y;         // 0..15

    for (int i = tid; i < 3 * 2304; i += 128) {
        int n = i / 2304, r = i % 2304;
        int co = r / 144, r2 = r % 144;
        int ci = r2 / 9, tap = r2 % 9;
        wl[n * 2304 + (((ci >> 1) * 9 + tap) * 16 + co) * 2 + (ci & 1)] = wc[i];
    }
    for (int i = tid; i < 16 * 3 * 66; i += 128) {
        int ci = i / 198, r2 = i % 198;
        int kr = r2 / 66, col = r2 % 66;
        int gh = h + kr - 1;
        int gw = col - 1;
        float v = 0.0f;
        if ((unsigned)gh < 64u && (unsigned)gw < 64u)
            v = out0[(((long)b * 16 + ci) * 64 + gh) * 64 + gw] * 0.25f;
        xl[(((ci >> 1) * 3 + kr) * 68 + col) * 2 + (ci & 1)] = v;
    }
    __syncthreads();

    const int wave = tid >> 5;
    const int lane = tid & 31;
    const int pix  = lane & 15;
    const int half = lane >> 4;

    v8f acc[3] = {};
    #pragma unroll
    for (int kh = 0; kh < 3; ++kh)
    #pragma unroll
    for (int kw = 0; kw < 3; ++kw)
    #pragma unroll
    for (int q = 0; q < 4; ++q) {
        const int cp = 2 * q + half;
        const int widx = ((cp * 9 + kh * 3 + kw) * 16 + pix) * 2;
        const v2f bm = *(const v2f*)&xl[((cp * 3 + kh) * 68 + (wave * 16 + pix + kw)) * 2];
        #pragma unroll
        for (int n = 0; n < 3; ++n) {
            const v2f a = *(const v2f*)&wl[n * 2304 + widx];
            acc[n] = __builtin_amdgcn_wmma_f32_16x16x4_f32(
                         false, a, false, bm, (short)0, acc[n], false, false);
        }
    }

    const float ow0 = sigm(oww[0]);
    const float own[3] = { sigm(oww[1]), sigm(oww[2]), sigm(oww[3]) };
    const float sgn[3] = { sigm(taus[0]), sigm(taus[1]), sigm(taus[2]) };
    const int col = wave * 16 + pix;

    #pragma unroll
    for (int g = 0; g < 8; ++g) {
        const int co = g + 8 * half;
        // out0 at (co, pixel): center tap of LDS tile holds xin = out0/4
        const float o0 =
            xl[(((co >> 1) * 3 + 1) * 68 + (wave * 16 + pix + 1)) * 2 + (co & 1)] * 4.0f;
        float total = o0 * ow0;
        #pragma unroll
        for (int n = 0; n < 3; ++n) {
            const int bi = n * 16 + co;
            const float inv  = bng[bi] * rsqrtf(bnv[bi] + 1e-5f);
            const float bias = bnb[bi] - bnm[bi] * inv;
            const float y    = acc[n][g] * inv + bias;
            const long idx = ((((long)n * 16 + b) * 16 + co) * 64 + h) * 64 + col;
            const float vo = vi[idx];
            const float vn = vo + (y - vo) * sgn[n];
            const float s  = (vn >= 1.0f) ? 1.0f : 0.0f;
            vi[idx] = vn * (1.0f - s);
            total += s * own[n];
        }
        outt[(((long)b * 16 + co) * 64 + h) * 64 + col] = total;
    }
}

// ---------------------------------------------------------------------------
// Host launcher: 8 sequential timesteps (state carry), 3 kernels each.
// ---------------------------------------------------------------------------
extern "C" void kernel_launch(void* const* d_in, const int* in_sizes, int n_in,
                              void* d_out, int out_size, void* d_ws, size_t ws_size,
                              hipStream_t stream) {
    (void)in_sizes; (void)n_in; (void)out_size; (void)ws_size;
    const float* x     = (const float*)d_in[0];
    const float* w0    = (const float*)d_in[1];
    const float* bn0g  = (const float*)d_in[2];
    const float* bn0b  = (const float*)d_in[3];
    const float* bn0m  = (const float*)d_in[4];
    const float* bn0v  = (const float*)d_in[5];
    const float* tau0  = (const float*)d_in[6];
    const float* wcv   = (const float*)d_in[7];
    const float* bnGg  = (const float*)d_in[8];
    const float* bnGb  = (const float*)d_in[9];
    const float* bnGm  = (const float*)d_in[10];
    const float* bnGv  = (const float*)d_in[11];
    const float* taus  = (const float*)d_in[12];
    const float* oww   = (const float*)d_in[13];
    float* out = (float*)d_out;

    // Workspace layout (floats)
    float* ws  = (float*)d_ws;
    const long V0N = (long)BB * CO * HH * WW;        // 4,194,304
    const long VIN = (long)NM1 * BB * CO * H2 * W2;  // 3,145,728
    float* v0   = ws;
    float* vi   = ws + V0N;
    float* s0   = ws + V0N + VIN;
    float* o0   = ws + V0N + VIN + V0N;

    // Zero the LIF membrane states (contiguous at workspace start).
    hipMemsetAsync(d_ws, 0, (size_t)(V0N + VIN) * sizeof(float), stream);

    const long xStride   = (long)BB * CC * HH * WW;  // per-timestep input
    const long outStride = (long)BB * CO * H2 * W2;  // per-timestep output
    const int  poolN     = BB * CO * H2 * W2;

    for (int t = 0; t < TT; ++t) {
        conv0_lif_kernel<<<dim3(2, 128, 16), 128, 0, stream>>>(
            x + (long)t * xStride, w0, bn0g, bn0b, bn0m, bn0v, tau0, v0, s0);
        pool_kernel<<<(poolN + 255) / 256, 256, 0, stream>>>(s0, o0);
        inner_kernel<<<dim3(64, 16), 128, 0, stream>>>(
            o0, wcv, bnGg, bnGb, bnGm, bnGv, taus, oww,
            vi, out + (long)t * outStride);
    }
}